// Stochastic3DKNNSmoothnessLoss_85933705658812
// MI455X (gfx1250) — compile-verified
//
#include <hip/hip_runtime.h>
#include <hip/hip_bf16.h>

typedef __attribute__((ext_vector_type(2))) float v2f;
typedef __attribute__((ext_vector_type(8))) float v8f;

#define SAMPLE 8192
#define KNN    8
#define CHUNK  512   // columns staged in LDS per pass
#define EPSF   1e-12f

// branchless sorted insert (hot loop): pure min/max chain, no phi copies
__device__ __forceinline__ void insert9_nb(unsigned int (&k)[9], unsigned int key) {
    unsigned int c = key;
#pragma unroll
    for (int s = 0; s < 9; ++s) {
        unsigned int o  = k[s];
        unsigned int lo = c < o ? c : o;
        unsigned int hi = c < o ? o : c;
        k[s] = lo;
        c    = hi;
    }
}

// guarded sorted insert (merge phase: low take-rate, skip pays off)
__device__ __forceinline__ void insert9(unsigned int (&k)[9], unsigned int key) {
    if (key < k[8]) insert9_nb(k, key);
}

// ---------------------------------------------------------------------------
// Kernel 1: gather means[idx] -> sm4 (xyz0), sq = |m|^2, sh0[idx] -> ss (rgb0)
// ---------------------------------------------------------------------------
__global__ __launch_bounds__(256) void knn_gather(
    const float* __restrict__ means, const float* __restrict__ sh0,
    const long long* __restrict__ idx,
    float* __restrict__ sm4, float* __restrict__ sq, float* __restrict__ ss,
    float* __restrict__ acc)
{
    int i = blockIdx.x * blockDim.x + threadIdx.x;
    if (i == 0) *acc = 0.0f;
    if (i < SAMPLE) {
        long long j = idx[i];
        float x = means[j * 3 + 0];
        float y = means[j * 3 + 1];
        float z = means[j * 3 + 2];
        sm4[i * 4 + 0] = x;
        sm4[i * 4 + 1] = y;
        sm4[i * 4 + 2] = z;
        sm4[i * 4 + 3] = 0.0f;
        sq[i] = x * x + y * y + z * z;
        ss[i * 4 + 0] = sh0[j * 3 + 0];
        ss[i * 4 + 1] = sh0[j * 3 + 1];
        ss[i * 4 + 2] = sh0[j * 3 + 2];
        ss[i * 4 + 3] = 0.0f;
    }
}

// ---------------------------------------------------------------------------
// Kernel 2: WMMA pairwise d^2 + per-lane packed top-9 + merge + loss
// Block = 128 threads (4 waves); each wave owns 16 rows -> 64 rows/block.
// ---------------------------------------------------------------------------
__global__ __launch_bounds__(128) void knn_main(
    const float* __restrict__ sm4, const float* __restrict__ sq,
    const float* __restrict__ ss, float* __restrict__ acc)
{
    // 4 waves * 16 rows * 16 lane-classes * 9 keys = 9216 u32 = 36 KB
    __shared__ __attribute__((aligned(16))) unsigned int smem[4 * 16 * 16 * 9];
    float* s_cm = (float*)smem;          // staging: 512 cols * float4
    float* s_sq = (float*)(smem + 2048); // staging: 512 sq values

    const int lane  = threadIdx.x & 31;
    const int wave  = threadIdx.x >> 5;
    const int hi    = lane >> 4;          // 0 = lanes 0-15, 1 = lanes 16-31
    const int ncls  = lane & 15;          // N (column) class within 16-tile
    const int kb    = hi * 2;             // K base for A/B operands
    const int rowBase = blockIdx.x * 64 + wave * 16;

    // A operand: 16x4 f32, lane L: M = L%16, K = kb..kb+1 (K=3 slot is zero)
    v2f a;
    a.x = sm4[(rowBase + ncls) * 4 + kb + 0];
    a.y = sm4[(rowBase + ncls) * 4 + kb + 1];

    // |m|^2 for the 8 rows this lane sees in the C/D layout
    float sqm[8];
#pragma unroll
    for (int v = 0; v < 8; ++v) sqm[v] = sq[rowBase + v + 8 * hi];

    // packed top-9 keys per owned row (key = d2 bits | column index)
    unsigned int keys[8][9];
#pragma unroll
    for (int v = 0; v < 8; ++v)
#pragma unroll
        for (int s = 0; s < 9; ++s) keys[v][s] = 0xFFFFFFFFu;

    for (int cb = 0; cb < SAMPLE; cb += CHUNK) {
        __syncthreads();  // previous chunk fully consumed
        // async stage of CHUNK columns directly into LDS (no VGPR round-trip)
        for (int t = threadIdx.x; t < CHUNK; t += 128) {
            const int      c      = cb + t;
            const unsigned ldsCm  = (unsigned)(unsigned long long)&((float4*)s_cm)[t];
            const unsigned ldsSq  = (unsigned)(unsigned long long)&s_sq[t];
            const unsigned offCm  = (unsigned)c * 16u;
            const unsigned offSq  = (unsigned)c * 4u;
            asm volatile("global_load_async_to_lds_b128 %0, %1, %2 offset:0"
                         :: "v"(ldsCm), "v"(offCm), "s"(sm4) : "memory");
            asm volatile("global_load_async_to_lds_b32 %0, %1, %2 offset:0"
                         :: "v"(ldsSq), "v"(offSq), "s"(sq) : "memory");
        }
        // prefetch next chunk toward L2 while we compute this one
        if (cb + CHUNK < SAMPLE && threadIdx.x < 16)
            __builtin_prefetch(&sm4[(cb + CHUNK) * 4 + threadIdx.x * 128], 0, 1);
        asm volatile("s_wait_asynccnt 0" ::: "memory");
        __syncthreads();

#pragma unroll 2
        for (int ct = 0; ct < CHUNK / 16; ++ct) {
            const int cl = ct * 16 + ncls;
            // B operand: 4x16 f32, lane L: N = L%16, K = kb..kb+1
            v2f b;
            b.x = s_cm[cl * 4 + kb + 0];
            b.y = s_cm[cl * 4 + kb + 1];
            const float sqn = s_sq[cl];

            v8f c0 = {};
            v8f dot = __builtin_amdgcn_wmma_f32_16x16x4_f32(
                false, a, false, b, (short)0, c0, false, false);

            const unsigned int jcol = (unsigned int)(cb + cl);
#pragma unroll
            for (int v = 0; v < 8; ++v) {
                // select on d^2 (monotonic with dist); clamp keeps key positive
                // and preserves the self/duplicate candidates (d2 ~ 0 or < 0)
                float d2 = fmaxf(__builtin_fmaf(-2.0f, dot[v], sqm[v] + sqn),
                                 EPSF);
                unsigned int key =
                    (__float_as_uint(d2) & 0xFFFFE000u) | jcol;
                insert9_nb(keys[v], key);   // branchless
            }
        }
    }

    __syncthreads();
    // dump per-lane lists for merge: [wave][row][ncls][9]
#pragma unroll
    for (int v = 0; v < 8; ++v) {
        int m = v + 8 * hi;
        unsigned int* dst = &smem[((wave * 16 + m) * 16 + ncls) * 9];
#pragma unroll
        for (int s = 0; s < 9; ++s) dst[s] = keys[v][s];
    }
    __syncthreads();

    // lanes 0..15: merge 16 partial lists -> global top-9 for one row, loss
    if (lane < 16) {
        const int i = rowBase + lane;
        unsigned int best[9];
#pragma unroll
        for (int s = 0; s < 9; ++s) best[s] = 0xFFFFFFFFu;
        const unsigned int* src = &smem[(wave * 16 + lane) * 144];
        for (int t = 0; t < 144; ++t) insert9(best, src[t]);

        const float mix = sm4[i * 4 + 0], miy = sm4[i * 4 + 1],
                    miz = sm4[i * 4 + 2];
        const float sqi = sq[i];
        const float s0x = ss[i * 4 + 0], s0y = ss[i * 4 + 1],
                    s0z = ss[i * 4 + 2];

        float lsum = 0.0f;
#pragma unroll
        for (int t = 1; t <= KNN; ++t) {  // best[0] == self, drop it
            int   j  = (int)(best[t] & 0x1FFFu);
            float d2 = sqi + sq[j] -
                       2.0f * (mix * sm4[j * 4 + 0] + miy * sm4[j * 4 + 1] +
                               miz * sm4[j * 4 + 2]);
            float dist = __builtin_sqrtf(fmaxf(d2, EPSF));  // exact, 8x per row
            float w    = expf(-dist);
            float dx = s0x - ss[j * 4 + 0];
            float dy = s0y - ss[j * 4 + 1];
            float dz = s0z - ss[j * 4 + 2];
            lsum += w * (dx * dx + dy * dy + dz * dz);
        }
        atomicAdd(acc, lsum);
    }
}

// ---------------------------------------------------------------------------
// Kernel 3: finalize mean
// ---------------------------------------------------------------------------
__global__ void knn_finalize(const float* __restrict__ acc,
                             float* __restrict__ out)
{
    out[0] = acc[0] * (1.0f / (8192.0f * 8.0f * 3.0f));
}

extern "C" void kernel_launch(void* const* d_in, const int* in_sizes, int n_in,
                              void* d_out, int out_size, void* d_ws, size_t ws_size,
                              hipStream_t stream)
{
    const float*     means = (const float*)d_in[0];
    const float*     sh0   = (const float*)d_in[1];
    const long long* idx   = (const long long*)d_in[2];

    float* ws  = (float*)d_ws;
    float* sm4 = ws;                       // 8192*4
    float* sq  = sm4 + SAMPLE * 4;         // 8192
    float* ss  = sq + SAMPLE;              // 8192*4
    float* acc = ss + SAMPLE * 4;          // 1

    knn_gather<<<(SAMPLE + 255) / 256, 256, 0, stream>>>(means, sh0, idx,
                                                         sm4, sq, ss, acc);
    knn_main<<<SAMPLE / 64, 128, 0, stream>>>(sm4, sq, ss, acc);
    knn_finalize<<<1, 1, 0, stream>>>(acc, (float*)d_out);
}